// ASLGNN_16836271800378
// MI455X (gfx1250) — compile-verified
//
#include <hip/hip_runtime.h>

#define N_NODES 840000
#define N_EDGES 1280000
#define N_GRAPHS 40000
#define NPG 21
#define HID 64
#define NCLS 29

typedef __attribute__((ext_vector_type(2))) float v2f;
typedef __attribute__((ext_vector_type(8))) float v8f;

// ---------------- degree / normalization ----------------
__global__ void k_deg_init(float* __restrict__ deg) {
    int i = blockIdx.x * 256 + threadIdx.x;
    if (i < N_NODES) deg[i] = 1.0f;           // self-loop
}

__global__ void k_deg_edges(const long long* __restrict__ dst, float* __restrict__ deg) {
    int e = blockIdx.x * 256 + threadIdx.x;
    if (e < N_EDGES) atomicAdd(&deg[(int)dst[e]], 1.0f);
}

__global__ void k_rsqrt(float* __restrict__ dinv) {
    int i = blockIdx.x * 256 + threadIdx.x;
    if (i < N_NODES) dinv[i] = rsqrtf(dinv[i]);
}

// ---------------- layer-1 linear (K=3), pre-scaled by dinv[node] ----------------
__global__ void k_lin1(const float* __restrict__ x, const float* __restrict__ W1,
                       const float* __restrict__ dinv, float* __restrict__ g) {
    int idx = blockIdx.x * 256 + threadIdx.x;
    if (idx >= N_NODES * HID) return;
    int i = idx >> 6, f = idx & 63;
    float h = x[i * 3 + 0] * W1[0 * HID + f]
            + x[i * 3 + 1] * W1[1 * HID + f]
            + x[i * 3 + 2] * W1[2 * HID + f];
    g[idx] = dinv[i] * h;
}

// ---------------- edge scatter: acc[dst] += g[src]  (acc pre-seeded with g = self-loop) ----
__global__ void k_scatter(const long long* __restrict__ src, const long long* __restrict__ dst,
                          const float* __restrict__ g, float* __restrict__ acc) {
    long long idx = (long long)blockIdx.x * 256 + threadIdx.x;
    if (idx >= (long long)N_EDGES * HID) return;
    int e = (int)(idx >> 6), f = (int)(idx & 63);
    int s = (int)src[e], d = (int)dst[e];
    atomicAdd(&acc[(size_t)d * HID + f], g[(size_t)s * HID + f]);
}

// ---------------- post: h = relu(dinv[i]*acc + bias[f]) in place ----------------
__global__ void k_post(const float* __restrict__ dinv, const float* __restrict__ bias,
                       float* __restrict__ acc) {
    int idx = blockIdx.x * 256 + threadIdx.x;
    if (idx >= N_NODES * HID) return;
    int i = idx >> 6, f = idx & 63;
    acc[idx] = fmaxf(fmaf(dinv[i], acc[idx], bias[f]), 0.0f);
}

// ---------------- layer-2 GEMM: D = dinv ⊙ (A[840k,64] @ B[64,64]) via fp32 WMMA ---------
// one wave per 16x16 output tile; K=64 -> 16 chained V_WMMA_F32_16X16X4_F32
__global__ void __launch_bounds__(256) k_gemm_wmma(const float* __restrict__ A,
                                                   const float* __restrict__ B,
                                                   const float* __restrict__ dinv,
                                                   float* __restrict__ D) {
    int wave = (blockIdx.x * 256 + threadIdx.x) >> 5;   // 210000 waves, exact
    int lane = threadIdx.x & 31;
    int rowTile = wave >> 2;                            // 52500 row tiles
    int colTile = wave & 3;                             // 4 col tiles of 16
    int half = lane >> 4, l16 = lane & 15;

    const float* arow = A + (size_t)(rowTile * 16 + l16) * HID;   // A row for this lane
    const float* bcol = B + colTile * 16 + l16;                   // B column for this lane
    v8f c = {};
#pragma unroll
    for (int k0 = 0; k0 < HID; k0 += 4) {
        int k = k0 + 2 * half;                 // lanes 0-15: K={k0,k0+1}; 16-31: {k0+2,k0+3}
        v2f a = *(const v2f*)(arow + k);       // 8B-aligned pair load
        v2f b;
        b.x = bcol[(size_t)k * HID];
        b.y = bcol[(size_t)(k + 1) * HID];
        c = __builtin_amdgcn_wmma_f32_16x16x4_f32(false, a, false, b, (short)0, c, false, false);
    }
    int col = colTile * 16 + l16;
#pragma unroll
    for (int j = 0; j < 8; ++j) {              // VGPR j holds rows j (lanes 0-15), j+8 (16-31)
        int row = rowTile * 16 + j + 8 * half;
        D[(size_t)row * HID + col] = c[j] * dinv[row];
    }
}

// ---------------- pool: relu(dinv*acc + b2) then mean over 21 nodes/graph ----------------
__global__ void k_pool(const float* __restrict__ dinv, const float* __restrict__ b2,
                       const float* __restrict__ acc, float* __restrict__ pooled) {
    int idx = blockIdx.x * 256 + threadIdx.x;
    if (idx >= N_GRAPHS * HID) return;
    int g = idx >> 6, f = idx & 63;
    float bias = b2[f];
    float s = 0.0f;
    int base = g * NPG;
#pragma unroll
    for (int j = 0; j < NPG; ++j) {
        int node = base + j;
        s += fmaxf(fmaf(dinv[node], acc[(size_t)node * HID + f], bias), 0.0f);
    }
    pooled[idx] = s * (1.0f / NPG);
}

// ---------------- FC head: out[40k,29] = pooled @ Wfc + bfc via fp32 WMMA (N padded to 32)
__global__ void __launch_bounds__(256) k_fc_wmma(const float* __restrict__ P,
                                                 const float* __restrict__ Wfc,
                                                 const float* __restrict__ bfc,
                                                 float* __restrict__ out) {
    int wave = (blockIdx.x * 256 + threadIdx.x) >> 5;   // 5000 waves, exact
    int lane = threadIdx.x & 31;
    int rowTile = wave >> 1;                            // 2500 row tiles
    int colTile = wave & 1;                             // cols [0,16) / [16,32)
    int half = lane >> 4, l16 = lane & 15;
    int col = colTile * 16 + l16;
    bool valid = (col < NCLS);

    const float* arow = P + (size_t)(rowTile * 16 + l16) * HID;
    v8f c = {};
#pragma unroll
    for (int k0 = 0; k0 < HID; k0 += 4) {
        int k = k0 + 2 * half;
        v2f a = *(const v2f*)(arow + k);
        v2f b;
        b.x = valid ? Wfc[(size_t)k * NCLS + col] : 0.0f;        // cndmask, no EXEC change
        b.y = valid ? Wfc[(size_t)(k + 1) * NCLS + col] : 0.0f;
        c = __builtin_amdgcn_wmma_f32_16x16x4_f32(false, a, false, b, (short)0, c, false, false);
    }
    if (valid) {
        float bb = bfc[col];
#pragma unroll
        for (int j = 0; j < 8; ++j) {
            int row = rowTile * 16 + j + 8 * half;
            out[(size_t)row * NCLS + col] = c[j] + bb;
        }
    }
}

extern "C" void kernel_launch(void* const* d_in, const int* in_sizes, int n_in,
                              void* d_out, int out_size, void* d_ws, size_t ws_size,
                              hipStream_t stream) {
    (void)in_sizes; (void)n_in; (void)out_size; (void)ws_size;
    const float*     x   = (const float*)d_in[0];
    const long long* ei  = (const long long*)d_in[1];   // [2, E] int64
    /* d_in[2] = batch (unused: graphs are 21 contiguous nodes) */
    const float*     W1  = (const float*)d_in[3];
    const float*     b1  = (const float*)d_in[4];
    const float*     W2  = (const float*)d_in[5];
    const float*     b2  = (const float*)d_in[6];
    const float*     Wfc = (const float*)d_in[7];
    const float*     bfc = (const float*)d_in[8];
    float* out = (float*)d_out;

    float* ws     = (float*)d_ws;
    float* dinv   = ws;                                  // [N]
    float* bufA   = dinv + N_NODES;                      // [N,64]
    float* bufB   = bufA + (size_t)N_NODES * HID;        // [N,64]
    float* pooled = bufB + (size_t)N_NODES * HID;        // [40k,64]

    const long long* src = ei;
    const long long* dst = ei + N_EDGES;

    const size_t featBytes = (size_t)N_NODES * HID * sizeof(float);
    const int nfBlocks = (N_NODES * HID + 255) / 256;
    const long long EW = (long long)N_EDGES * HID;
    const int ewBlocks = (int)((EW + 255) / 256);

    // degree -> dinv
    k_deg_init <<<(N_NODES + 255) / 256, 256, 0, stream>>>(dinv);
    k_deg_edges<<<(N_EDGES + 255) / 256, 256, 0, stream>>>(dst, dinv);
    k_rsqrt    <<<(N_NODES + 255) / 256, 256, 0, stream>>>(dinv);

    // layer 1
    k_lin1<<<nfBlocks, 256, 0, stream>>>(x, W1, dinv, bufA);                  // g1 = dinv*(x@W1)
    hipMemcpyAsync(bufB, bufA, featBytes, hipMemcpyDeviceToDevice, stream);   // acc = g1 (self-loop)
    k_scatter<<<ewBlocks, 256, 0, stream>>>(src, dst, bufA, bufB);            // acc += g1[src]
    k_post<<<nfBlocks, 256, 0, stream>>>(dinv, b1, bufB);                     // h1 = relu(dinv*acc+b1)

    // layer 2 (WMMA fp32 GEMM)
    k_gemm_wmma<<<(N_NODES / 16) * 4 / 8, 256, 0, stream>>>(bufB, W2, dinv, bufA); // g2
    hipMemcpyAsync(bufB, bufA, featBytes, hipMemcpyDeviceToDevice, stream);   // acc2 = g2
    k_scatter<<<ewBlocks, 256, 0, stream>>>(src, dst, bufA, bufB);            // acc2 += g2[src]

    // pool + FC head (WMMA fp32)
    k_pool<<<(N_GRAPHS * HID + 255) / 256, 256, 0, stream>>>(dinv, b2, bufB, pooled);
    k_fc_wmma<<<(N_GRAPHS / 16) * 2 / 8, 256, 0, stream>>>(pooled, Wfc, bfc, out);
}